// attention_feed_25082609008750
// MI455X (gfx1250) — compile-verified
//
#include <hip/hip_runtime.h>
#include <math.h>

// Problem constants (match reference)
#define BB 32
#define SS 2048
#define DD 1024
#define MASK_PENALTY 100000000.0f

typedef __attribute__((ext_vector_type(2))) float v2f;
typedef __attribute__((ext_vector_type(8))) float v8f;

// ---------------------------------------------------------------------------
// Kernel 0: zero the output (needed for atomic accumulation in combine pass)
// ---------------------------------------------------------------------------
__global__ void af_zero_kernel(float* __restrict__ out, int n) {
    int i = blockIdx.x * blockDim.x + threadIdx.x;
    if (i < n) out[i] = 0.0f;
}

// ---------------------------------------------------------------------------
// Kernel 1: scores[b,s] = dot(context[b,s,:], target[b,:]) via WMMA f32 16x16x4
//
// grid = (B, S/128), block = 256 threads (8 waves). Each wave computes 16
// scores. A-fragment (16x4 f32): lane&15 selects row M, lane>>4 selects the
// K-pair (K=0,1 in low half / K=2,3 in high half) -> each lane streams a
// context row as float2 with stride 16 bytes; lanes l and l+16 share a row
// and cover complementary float2s, so every cacheline is fully used.
// B-fragment: target values placed only in column N==0 (lanes 0 and 16);
// column 0 of the accumulator then holds the 16 dot products.
// ---------------------------------------------------------------------------
__global__ void af_scores_wmma_kernel(const float* __restrict__ tgt,
                                      const float* __restrict__ ctx,
                                      float* __restrict__ scores) {
    const int b     = blockIdx.x;            // 0..31
    const int chunk = blockIdx.y;            // 0..15  (128 s-rows each)
    const int wave  = threadIdx.x >> 5;      // 0..7
    const int lane  = threadIdx.x & 31;

    const int s0   = chunk * 128 + wave * 16;       // first s-row of this wave
    const int rowA = s0 + (lane & 15);              // M index -> s row
    const int kofs = (lane >> 4) * 2;               // 0 for low half, 2 for high

    const float* __restrict__ ap = ctx + ((size_t)b * SS + rowA) * DD + kofs;
    const float* __restrict__ tp = tgt + (size_t)b * DD + kofs;

    // keep only column N==0 of the B matrix
    const float bsel = ((lane & 15) == 0) ? 1.0f : 0.0f;

    v8f c = {0.f, 0.f, 0.f, 0.f, 0.f, 0.f, 0.f, 0.f};

#pragma unroll 8
    for (int k = 0; k < DD; k += 4) {
        v2f a  = *(const v2f*)(ap + k);      // A: ctx[rowA, k+kofs .. k+kofs+1]
        v2f tb = *(const v2f*)(tp + k);      // broadcast load of target pair
        v2f bmat;
        bmat.x = tb.x * bsel;                // B[K=kofs  , N=0] = t[k+kofs]
        bmat.y = tb.y * bsel;                // B[K=kofs+1, N=0] = t[k+kofs+1]
        // D = A(16x4) * B(4x16) + C  ; emits v_wmma_f32_16x16x4_f32
        c = __builtin_amdgcn_wmma_f32_16x16x4_f32(
                /*neg_a=*/false, a, /*neg_b=*/false, bmat,
                /*c_mod=*/(short)0, c, /*reuse_a=*/false, /*reuse_b=*/false);
    }

    // Column N=0 lives in lane 0 (M = 0..7 in c[0..7]) and lane 16 (M = 8..15).
    if ((lane & 15) == 0) {
        const int base = s0 + (lane >> 4) * 8;
        float* __restrict__ sp = scores + (size_t)b * SS + base;
#pragma unroll
        for (int v = 0; v < 8; ++v) sp[v] = c[v];
    }
}

// ---------------------------------------------------------------------------
// Kernel 2: masked softmax over S per batch row. grid = B, block = 256.
// weights written back in place into the scores buffer.
// ---------------------------------------------------------------------------
__global__ void af_softmax_kernel(const float* __restrict__ mask,
                                  float* __restrict__ scores) {
    const int b = blockIdx.x;
    const int t = threadIdx.x;               // 0..255, 8 elements per thread
    __shared__ float red[256];

    float vals[8];
    float m = -INFINITY;
#pragma unroll
    for (int i = 0; i < 8; ++i) {
        const int s = t + i * 256;
        float v = scores[(size_t)b * SS + s]
                + (mask[(size_t)b * SS + s] - 1.0f) * MASK_PENALTY;
        vals[i] = v;
        m = fmaxf(m, v);
    }

    red[t] = m;
    __syncthreads();
    for (int off = 128; off > 0; off >>= 1) {
        if (t < off) red[t] = fmaxf(red[t], red[t + off]);
        __syncthreads();
    }
    m = red[0];
    __syncthreads();

    float sum = 0.0f;
#pragma unroll
    for (int i = 0; i < 8; ++i) {
        vals[i] = expf(vals[i] - m);
        sum += vals[i];
    }
    red[t] = sum;
    __syncthreads();
    for (int off = 128; off > 0; off >>= 1) {
        if (t < off) red[t] += red[t + off];
        __syncthreads();
    }
    const float inv = 1.0f / red[0];

#pragma unroll
    for (int i = 0; i < 8; ++i)
        scores[(size_t)b * SS + t + i * 256] = vals[i] * inv;
}

// ---------------------------------------------------------------------------
// Kernel 3: out[b,d] = sum_s w[b,s] * ctx[b,s,d], split over S with atomics.
// grid = (B, S/128), block = 256 threads; thread t owns columns 4t..4t+3,
// streams 128 full rows with b128 loads (weights are wave-uniform -> s_load).
// ---------------------------------------------------------------------------
__global__ void af_combine_kernel(const float* __restrict__ ctx,
                                  const float* __restrict__ w,
                                  float* __restrict__ out) {
    const int b     = blockIdx.x;
    const int chunk = blockIdx.y;            // 0..15
    const int t     = threadIdx.x;           // 0..255
    const int d0    = t * 4;

    const float* __restrict__ cp = ctx + ((size_t)b * SS + (size_t)chunk * 128) * DD + d0;
    const float* __restrict__ wp = w + (size_t)b * SS + (size_t)chunk * 128;

    const long rows_left_global = (long)BB * SS - ((long)b * SS + chunk * 128);

    float4 acc = make_float4(0.f, 0.f, 0.f, 0.f);
#pragma unroll 4
    for (int s = 0; s < 128; ++s) {
        if ((s & 7) == 0 && (s + 8) < rows_left_global) {
            // CDNA5 prefetch path: emits global_prefetch_b8
            __builtin_prefetch(cp + (size_t)(s + 8) * DD, 0, 1);
        }
        const float ww = wp[s];              // uniform -> scalar load
        const float4 v = *(const float4*)(cp + (size_t)s * DD);
        acc.x = fmaf(ww, v.x, acc.x);
        acc.y = fmaf(ww, v.y, acc.y);
        acc.z = fmaf(ww, v.z, acc.z);
        acc.w = fmaf(ww, v.w, acc.w);
    }

    float* __restrict__ op = out + (size_t)b * DD + d0;
    atomicAdd(op + 0, acc.x);
    atomicAdd(op + 1, acc.y);
    atomicAdd(op + 2, acc.z);
    atomicAdd(op + 3, acc.w);
}

// ---------------------------------------------------------------------------
// Launcher
// ---------------------------------------------------------------------------
extern "C" void kernel_launch(void* const* d_in, const int* in_sizes, int n_in,
                              void* d_out, int out_size, void* d_ws, size_t ws_size,
                              hipStream_t stream) {
    const float* tgt  = (const float*)d_in[0];   // [B, D]
    const float* ctx  = (const float*)d_in[1];   // [B, S, D]
    const float* mask = (const float*)d_in[2];   // [B, S]
    float* out    = (float*)d_out;               // [B, D]
    float* scores = (float*)d_ws;                // [B, S] scratch (256 KB)

    // 1) zero output for atomic accumulation
    af_zero_kernel<<<(BB * DD + 255) / 256, 256, 0, stream>>>(out, BB * DD);

    // 2) scores = context . target  (WMMA f32 16x16x4)
    dim3 gs(BB, SS / 128);
    af_scores_wmma_kernel<<<gs, 256, 0, stream>>>(tgt, ctx, scores);

    // 3) masked softmax over S (in place)
    af_softmax_kernel<<<BB, 256, 0, stream>>>(mask, scores);

    // 4) out = weights . context (split-S with fp32 atomics)
    dim3 gc(BB, SS / 128);
    af_combine_kernel<<<gc, 256, 0, stream>>>(ctx, scores, out);
}